// Superattention_conv_79903571575312
// MI455X (gfx1250) — compile-verified
//
#include <hip/hip_runtime.h>
#include <hip/hip_bf16.h>
#include <math.h>

typedef float v2f __attribute__((ext_vector_type(2)));
typedef float v8f __attribute__((ext_vector_type(8)));

#define B_   16
#define C_   128
#define S_   2048
#define HW_  4096
#define LSTR 130          // padded LDS row stride in floats (even -> b64 aligned, 2*lane banks)
#define EPSN 1e-5f
#define INV_TEMP 100.0f

// ---------------------------------------------------------------------------
// K1: per-batch ragged lengths from label maps: len = max(label) + 1
// ---------------------------------------------------------------------------
__global__ void __launch_bounds__(256) len_kernel(const int* __restrict__ lmt,
                                                  const int* __restrict__ lmr,
                                                  int* __restrict__ lens) {
  __shared__ int red[256];
  const int b = blockIdx.x;
  const int* src = ((blockIdx.y == 0) ? lmt : lmr) + b * HW_;
  int m = -2147483647 - 1;
  for (int i = threadIdx.x; i < HW_; i += 256) m = max(m, src[i]);
  red[threadIdx.x] = m;
  __syncthreads();
  for (int s = 128; s > 0; s >>= 1) {
    if (threadIdx.x < (unsigned)s) red[threadIdx.x] = max(red[threadIdx.x], red[threadIdx.x + s]);
    __syncthreads();
  }
  if (threadIdx.x == 0) lens[blockIdx.y * B_ + b] = red[0] + 1;
}

// ---------------------------------------------------------------------------
// K2: fused center-norm (optional) + 1x1 conv (C x C matmul) for a 128-seg tile
//   Y[b][o][s] = sum_c W[o][c] * Xn[b][c][s] + bias[o]
// ---------------------------------------------------------------------------
__global__ void __launch_bounds__(256) qkv_kernel(const float* __restrict__ X,
                                                  const float* __restrict__ W,
                                                  const float* __restrict__ bias,
                                                  float* __restrict__ Y,
                                                  int do_norm) {
  __shared__ float Xt[128 * LSTR];  // [seg][chan] transposed slab
  const int b  = blockIdx.y;
  const int s0 = blockIdx.x * 128;
  const int tid = threadIdx.x;
  const float* Xb = X + (size_t)b * C_ * S_;

  for (int idx = tid; idx < 128 * 128; idx += 256) {
    const int c = idx >> 7, s = idx & 127;          // s fastest -> coalesced global
    Xt[s * LSTR + c] = Xb[(size_t)c * S_ + s0 + s];
  }
  __syncthreads();

  if (do_norm && tid < 128) {
    float* row = &Xt[tid * LSTR];
    float sum = 0.f;
    for (int c = 0; c < C_; ++c) sum += row[c];
    const float mean = sum * (1.0f / C_);
    float ss = 0.f;
    for (int c = 0; c < C_; ++c) { const float d = row[c] - mean; ss += d * d; }
    const float r = 1.0f / (sqrtf(ss) + EPSN);
    for (int c = 0; c < C_; ++c) row[c] = (row[c] - mean) * r;
  }
  __syncthreads();

  const int wave = tid >> 5;
  const int lane = tid & 31;
  const int lm = lane & 15;
  const int kb = (lane & 16) ? 2 : 0;
  const int o0 = wave * 16;

  v2f a[32];
  const float* wrow = W + (o0 + lm) * C_ + kb;      // A[M][K] contiguous float2
  #pragma unroll
  for (int kk = 0; kk < 32; ++kk) a[kk] = *(const v2f*)(wrow + kk * 4);

  float* Yb = Y + (size_t)b * C_ * S_;
  const int rbase = o0 + ((lane & 16) ? 8 : 0);
  #pragma unroll 1
  for (int j = 0; j < 8; ++j) {
    v8f acc = {0.f, 0.f, 0.f, 0.f, 0.f, 0.f, 0.f, 0.f};
    #pragma unroll
    for (int kk = 0; kk < 32; ++kk) {
      const v2f bf = *(const v2f*)&Xt[(j * 16 + lm) * LSTR + kk * 4 + kb];
      acc = __builtin_amdgcn_wmma_f32_16x16x4_f32(false, a[kk], false, bf,
                                                  (short)0, acc, false, false);
    }
    const int col = s0 + j * 16 + lm;
    #pragma unroll
    for (int r = 0; r < 8; ++r)
      Yb[(size_t)(rbase + r) * S_ + col] = acc[r] + bias[rbase + r];
  }
}

// ---------------------------------------------------------------------------
// K3: sim logits (no temperature yet): att[b][n][m] = sum_c q[b][c][n]*k[b][c][m]
// ---------------------------------------------------------------------------
__global__ void __launch_bounds__(256) sim_kernel(const float* __restrict__ q,
                                                  const float* __restrict__ k,
                                                  float* __restrict__ att) {
  __shared__ float Qt[128 * LSTR];   // [n][c]
  __shared__ float Kt[128 * LSTR];   // [m][c]
  const int b  = blockIdx.z;
  const int n0 = blockIdx.y * 128;
  const int m0 = blockIdx.x * 128;
  const int tid = threadIdx.x;
  const float* qb = q + (size_t)b * C_ * S_;
  const float* kb_ = k + (size_t)b * C_ * S_;

  for (int idx = tid; idx < 128 * 128; idx += 256) {
    const int c = idx >> 7, s = idx & 127;
    Qt[s * LSTR + c] = qb[(size_t)c * S_ + n0 + s];
    Kt[s * LSTR + c] = kb_[(size_t)c * S_ + m0 + s];
  }
  __syncthreads();

  const int wave = tid >> 5;
  const int lane = tid & 31;
  const int lm = lane & 15;
  const int kbo = (lane & 16) ? 2 : 0;

  v2f a[32];
  #pragma unroll
  for (int kk = 0; kk < 32; ++kk)
    a[kk] = *(const v2f*)&Qt[(wave * 16 + lm) * LSTR + kk * 4 + kbo];

  float* ab = att + (size_t)b * S_ * S_;
  const int rbase = n0 + wave * 16 + ((lane & 16) ? 8 : 0);
  #pragma unroll 1
  for (int j = 0; j < 8; ++j) {
    v8f acc = {0.f, 0.f, 0.f, 0.f, 0.f, 0.f, 0.f, 0.f};
    #pragma unroll
    for (int kk = 0; kk < 32; ++kk) {
      const v2f bf = *(const v2f*)&Kt[(j * 16 + lm) * LSTR + kk * 4 + kbo];
      acc = __builtin_amdgcn_wmma_f32_16x16x4_f32(false, a[kk], false, bf,
                                                  (short)0, acc, false, false);
    }
    const int col = m0 + j * 16 + lm;
    #pragma unroll
    for (int r = 0; r < 8; ++r)
      ab[(size_t)(rbase + r) * S_ + col] = acc[r];
  }
}

// ---------------------------------------------------------------------------
// K4: masked softmax over ref columns (scale INV_TEMP applied here), zero
//     invalid target rows. In-place on the att output buffer.
// ---------------------------------------------------------------------------
__global__ void __launch_bounds__(256) softmax_kernel(float* __restrict__ att,
                                                      const int* __restrict__ lens) {
  const int b = blockIdx.y;
  const int n = blockIdx.x;
  const int tid = threadIdx.x;
  const int len_t = lens[b];
  const int len_r = lens[B_ + b];
  float* row = att + (size_t)b * S_ * S_ + (size_t)n * S_;
  __shared__ float red[256];

  if (n >= len_t) {                        // invalid target row -> zeros
    for (int i = tid; i < S_; i += 256) row[i] = 0.0f;
    return;
  }

  float x[8];
  float mx = -3.402823466e38f;
  #pragma unroll
  for (int i = 0; i < 8; ++i) {
    const int m = tid + i * 256;
    x[i] = row[m];
    if (m < len_r) mx = fmaxf(mx, x[i]);
  }
  red[tid] = mx;
  __syncthreads();
  for (int s = 128; s > 0; s >>= 1) {
    if (tid < (unsigned)s) red[tid] = fmaxf(red[tid], red[tid + s]);
    __syncthreads();
  }
  mx = red[0];
  __syncthreads();

  float e[8];
  float sum = 0.f;
  #pragma unroll
  for (int i = 0; i < 8; ++i) {
    const int m = tid + i * 256;
    e[i] = (m < len_r) ? __expf(INV_TEMP * (x[i] - mx)) : 0.0f;
    sum += e[i];
  }
  red[tid] = sum;
  __syncthreads();
  for (int s = 128; s > 0; s >>= 1) {
    if (tid < (unsigned)s) red[tid] += red[tid + s];
    __syncthreads();
  }
  const float inv = 1.0f / red[0];
  #pragma unroll
  for (int i = 0; i < 8; ++i) row[tid + i * 256] = e[i] * inv;
}

// ---------------------------------------------------------------------------
// K5: out0[b][c][n] = sum_m att[b][n][m] * v[b][c][m]  (K = 2048, LDS-tiled att)
// ---------------------------------------------------------------------------
__global__ void __launch_bounds__(256) av_kernel(const float* __restrict__ att,
                                                 const float* __restrict__ v,
                                                 float* __restrict__ out) {
  __shared__ float At[128 * LSTR];   // [n][m] chunk, natural = B's [N][K] layout
  const int b  = blockIdx.y;
  const int n0 = blockIdx.x * 128;
  const int tid = threadIdx.x;
  const int wave = tid >> 5;
  const int lane = tid & 31;
  const int lm = lane & 15;
  const int kbo = (lane & 16) ? 2 : 0;
  const int c0 = wave * 16;
  const float* ab = att + (size_t)b * S_ * S_;
  const float* vb = v + (size_t)b * C_ * S_;

  v8f acc[8];
  #pragma unroll
  for (int j = 0; j < 8; ++j) acc[j] = (v8f){0.f, 0.f, 0.f, 0.f, 0.f, 0.f, 0.f, 0.f};

  for (int ch = 0; ch < 16; ++ch) {
    const int mb = ch * 128;
    __syncthreads();
    for (int idx = tid; idx < 128 * 128; idx += 256) {
      const int nl = idx >> 7, ml = idx & 127;
      At[nl * LSTR + ml] = ab[(size_t)(n0 + nl) * S_ + mb + ml];
    }
    __syncthreads();
    if (ch < 15)  // hint next att chunk into cache (global_prefetch_b8)
      __builtin_prefetch(&ab[(size_t)(n0 + (tid & 127)) * S_ + mb + 128], 0, 0);

    v2f a[32];
    const float* vrow = vb + (size_t)(c0 + lm) * S_ + mb + kbo;
    #pragma unroll
    for (int kk = 0; kk < 32; ++kk) a[kk] = *(const v2f*)(vrow + kk * 4);

    #pragma unroll
    for (int j = 0; j < 8; ++j) {
      #pragma unroll
      for (int kk = 0; kk < 32; ++kk) {
        const v2f bf = *(const v2f*)&At[(j * 16 + lm) * LSTR + kk * 4 + kbo];
        acc[j] = __builtin_amdgcn_wmma_f32_16x16x4_f32(false, a[kk], false, bf,
                                                       (short)0, acc[j], false, false);
      }
    }
  }

  float* ob = out + (size_t)b * C_ * S_;
  const int rbase = c0 + ((lane & 16) ? 8 : 0);
  #pragma unroll
  for (int j = 0; j < 8; ++j) {
    const int col = n0 + j * 16 + lm;
    #pragma unroll
    for (int r = 0; r < 8; ++r)
      ob[(size_t)(rbase + r) * S_ + col] = acc[j][r];
  }
}

// ---------------------------------------------------------------------------
extern "C" void kernel_launch(void* const* d_in, const int* in_sizes, int n_in,
                              void* d_out, int out_size, void* d_ws, size_t ws_size,
                              hipStream_t stream) {
  const float* tgt = (const float*)d_in[0];
  const float* ref = (const float*)d_in[1];
  const int*   lmt = (const int*)d_in[2];
  const int*   lmr = (const int*)d_in[3];
  const float* rgb = (const float*)d_in[6];
  const float* Wq  = (const float*)d_in[9];
  const float* bq  = (const float*)d_in[10];
  const float* Wk  = (const float*)d_in[11];
  const float* bk  = (const float*)d_in[12];
  const float* Wv  = (const float*)d_in[13];
  const float* bv  = (const float*)d_in[14];

  float* out0 = (float*)d_out;                           // [B, C, S] (wg transposed)
  float* att  = (float*)d_out + (size_t)B_ * C_ * S_;    // [B, S, S]

  char*  ws  = (char*)d_ws;
  int*   lens = (int*)ws;                                // [2][B]
  float* qws = (float*)(ws + 256);
  float* kws = qws + (size_t)B_ * C_ * S_;
  float* vws = kws + (size_t)B_ * C_ * S_;

  len_kernel<<<dim3(B_, 2), 256, 0, stream>>>(lmt, lmr, lens);
  qkv_kernel<<<dim3(S_ / 128, B_), 256, 0, stream>>>(tgt, Wq, bq, qws, 1);
  qkv_kernel<<<dim3(S_ / 128, B_), 256, 0, stream>>>(ref, Wk, bk, kws, 1);
  qkv_kernel<<<dim3(S_ / 128, B_), 256, 0, stream>>>(rgb, Wv, bv, vws, 0);
  sim_kernel<<<dim3(S_ / 128, S_ / 128, B_), 256, 0, stream>>>(qws, kws, att);
  softmax_kernel<<<dim3(S_, B_), 256, 0, stream>>>(att, lens);
  av_kernel<<<dim3(S_ / 128, B_), 256, 0, stream>>>(att, vws, out0);
}